// DWT2D_1185410974326
// MI455X (gfx1250) — compile-verified
//
#include <hip/hip_runtime.h>
#include <hip/hip_bf16.h>
#include <stdint.h>

// 2-D Haar DWT, one level, reflect-pad 1, stride 2.
//   x  : (B, C, 256, 256) f32
//   out: (B, 4C, 129, 129) f32, channel blocks [ll | lh | hl | hh]
//
// Bandwidth-bound: ~268 MB read + ~273 MB written => ~23 us floor @ 23.3 TB/s.
// CDNA5 paths used:
//   * global_load_async_to_lds_b128 (ASYNCcnt copy engine) for b128-coalesced
//     row staging into LDS, drained with s_wait_asynccnt.
//   * TH_NT streaming stores (__builtin_nontemporal_store) so the write
//     stream does not evict the read stream from L2.

#define H_IN   256
#define W_IN   256
#define OH     129
#define OW     129
#define ROWS_PB  8              // output rows per block
#define SLOTS   (2 * ROWS_PB)   // staged input rows per block

// padded index p in [0, H_IN+1] -> raw index, matching jnp reflect pad 1.
__device__ __forceinline__ int reflect_idx(int p) {
    int idx = p - 1;
    idx = (idx < 0) ? -idx : idx;                       // p==0   -> 1
    idx = (idx > (H_IN - 1)) ? (2 * (H_IN - 1) - idx)   // p==256 -> 255, 257 -> 254
                             : idx;
    return idx;
}

__global__ void __launch_bounds__(256)
haar_dwt2d_async(const float* __restrict__ x, float* __restrict__ out) {
    __shared__ float rows[SLOTS][W_IN];     // 16 KB staged input rows

    const int i0 = blockIdx.x * ROWS_PB;    // first output row of this block
    const int bc = blockIdx.y;              // fused (b, c) plane index
    const int t  = threadIdx.x;             // 0..255

    const float* src = x + (size_t)bc * (size_t)(H_IN * W_IN);

    // ---- async stage: 16 rows x 1 KB = 1024 b128 chunks, 4 per thread ------
#pragma unroll
    for (int iter = 0; iter < 4; ++iter) {
        const int k    = iter * 256 + t;    // chunk id 0..1023
        const int slot = k >> 6;            // staged row slot 0..15
        const int l16  = k & 63;            // 16B chunk within row
        const int i    = i0 + (slot >> 1);  // output row this slot serves
        if (i < OH) {
            const int      r     = reflect_idx(2 * i + (slot & 1));
            const uint64_t gaddr = (uint64_t)(uintptr_t)(src + (size_t)r * W_IN)
                                 + (uint64_t)l16 * 16u;
            const uint32_t laddr = (uint32_t)(uintptr_t)(&rows[slot][0])
                                 + (uint32_t)l16 * 16u; // low 32 bits = LDS offset
            asm volatile("global_load_async_to_lds_b128 %0, %1, off"
                         :
                         : "v"(laddr), "v"(gaddr)
                         : "memory");
        }
    }
    asm volatile("s_wait_asynccnt 0" ::: "memory");  // drain this wave's copies
    __syncthreads();                                  // visibility across waves

    // ---- compute: wave w -> output row i0+w; lane -> cols at stride 32 -----
    const int w    = t >> 5;                // wave 0..7
    const int lane = t & 31;
    const int i    = i0 + w;
    if (i >= OH) return;

    const float* __restrict__ r0 = rows[2 * w];
    const float* __restrict__ r1 = rows[2 * w + 1];

    const int    b       = bc >> 6;                          // C = 64
    const int    c       = bc & 63;
    const size_t plane   = (size_t)64 * OH * OW;             // C*OH*OW
    const size_t rowbase = ((size_t)b * 256 + (size_t)c) * (size_t)(OH * OW)
                         + (size_t)i * OW;

    // lo = [s, s], hi = [-s, s], s = 1/sqrt(2); outer products -> +/- 0.5
#pragma unroll
    for (int q = 0; q < 5; ++q) {
        const int j = lane + 32 * q;        // q==4 handles only col 128 (lane 0)
        if (j >= OW) break;

        const int c0 = reflect_idx(2 * j);
        const int c1 = reflect_idx(2 * j + 1);

        const float x00 = r0[c0], x01 = r0[c1];
        const float x10 = r1[c0], x11 = r1[c1];

        const float a0 = x00 + x01;   // row0 sum
        const float a1 = x10 + x11;   // row1 sum
        const float d0 = x01 - x00;   // row0 diff
        const float d1 = x11 - x10;   // row1 diff

        const size_t o = rowbase + (size_t)j;
        __builtin_nontemporal_store(0.5f * (a0 + a1), out + o);              // ll
        __builtin_nontemporal_store(0.5f * (d0 + d1), out + o + plane);      // lh
        __builtin_nontemporal_store(0.5f * (a1 - a0), out + o + 2 * plane);  // hl
        __builtin_nontemporal_store(0.5f * (d1 - d0), out + o + 3 * plane);  // hh
    }
}

extern "C" void kernel_launch(void* const* d_in, const int* in_sizes, int n_in,
                              void* d_out, int out_size, void* d_ws, size_t ws_size,
                              hipStream_t stream) {
    (void)n_in; (void)out_size; (void)d_ws; (void)ws_size;

    const float* x   = (const float*)d_in[0];
    float*       out = (float*)d_out;

    const int n_planes = in_sizes[0] / (H_IN * W_IN);   // B*C = 1024

    dim3 grid((OH + ROWS_PB - 1) / ROWS_PB, n_planes, 1);  // (17, 1024)
    dim3 block(256, 1, 1);                                  // 8 wave32s
    haar_dwt2d_async<<<grid, block, 0, stream>>>(x, out);
}